// LMHTNeuron_6837587935403
// MI455X (gfx1250) — compile-verified
//
#include <hip/hip_runtime.h>

// LIF-style integrate-and-fire with multi-level quantization.
//   x: (T=4, B=4, S=2048, D=2048) fp32, scale/zero_point: scalar fp32
//   aux = scale*zero_point/T ; v0 = 0.5 ; L = 64
// Pure streaming op: 512 MB traffic -> ~22us floor at 23.3 TB/s. Nonlinear
// temporal recurrence -> no GEMM structure -> WMMA inapplicable. Strategy:
// per-timestep base pointers passed as kernargs so ALL 8 b128 NT accesses
// lower to saddr + scale_offset form (SGPR64 base, one shared 32-bit VGPR
// index, zero per-access VALU address math); scan carried in registers.

typedef float v4f __attribute__((ext_vector_type(4)));

#define TSTEPS 4
#define QLEVELS 64.0f
#define V0 0.5f

__global__ __launch_bounds__(256) void lmht_lif_kernel(
    const v4f* __restrict__ x0, const v4f* __restrict__ x1,
    const v4f* __restrict__ x2, const v4f* __restrict__ x3,
    v4f* __restrict__ o0, v4f* __restrict__ o1,
    v4f* __restrict__ o2, v4f* __restrict__ o3,
    const float* __restrict__ scale_p,
    const float* __restrict__ zp_p,
    unsigned nvec)  // float4 vectors per timestep plane
{
    unsigned i = blockIdx.x * blockDim.x + threadIdx.x;
    if (i >= nvec) return;

    // Uniform scalars: scalar-cached loads, one division per thread.
    const float s   = scale_p[0];
    const float zp  = zp_p[0];
    const float inv = 1.0f / s;
    const float aux = s * zp * (1.0f / (float)TSTEPS);

    // All T loads in flight before first use; each is
    //   global_load_b128 v[..], v_i, s[base] scale_offset th:TH_LOAD_NT
    v4f xt[TSTEPS];
    xt[0] = __builtin_nontemporal_load(x0 + i);
    xt[1] = __builtin_nontemporal_load(x1 + i);
    xt[2] = __builtin_nontemporal_load(x2 + i);
    xt[3] = __builtin_nontemporal_load(x3 + i);

    v4f* const outp[TSTEPS] = {o0, o1, o2, o3};

    v4f v = {V0, V0, V0, V0};

#pragma unroll
    for (int t = 0; t < TSTEPS; ++t) {
        v4f res;
#pragma unroll
        for (int j = 0; j < 4; ++j) {
            float vv = v[j] + xt[t][j];                // integrate
            float k  = __builtin_floorf(vv * inv);     // quantize level
            k        = fminf(fmaxf(k, 0.0f), QLEVELS); // clamp -> v_med3
            float o  = k * s;                          // emitted charge
            v[j]     = vv - o;                         // soft reset
            res[j]   = o - aux;                        // spike output
        }
        // Streaming NT store (saddr + scale_offset form).
        __builtin_nontemporal_store(res, outp[t] + i);
    }
}

extern "C" void kernel_launch(void* const* d_in, const int* in_sizes, int n_in,
                              void* d_out, int out_size, void* d_ws, size_t ws_size,
                              hipStream_t stream) {
    const float* x     = (const float*)d_in[0];
    const float* scale = (const float*)d_in[1];
    const float* zp    = (const float*)d_in[2];
    float*       out   = (float*)d_out;

    const unsigned total = (unsigned)in_sizes[0];     // T*B*S*D
    const unsigned N     = total / (unsigned)TSTEPS;  // per-plane elements
    const unsigned nvec  = N / 4u;                    // float4 vectors / plane

    const v4f* x0 = (const v4f*)(x + 0ull * N);
    const v4f* x1 = (const v4f*)(x + 1ull * N);
    const v4f* x2 = (const v4f*)(x + 2ull * N);
    const v4f* x3 = (const v4f*)(x + 3ull * N);
    v4f* o0 = (v4f*)(out + 0ull * N);
    v4f* o1 = (v4f*)(out + 1ull * N);
    v4f* o2 = (v4f*)(out + 2ull * N);
    v4f* o3 = (v4f*)(out + 3ull * N);

    const int block = 256;  // 8 wave32 per block
    const unsigned grid = (nvec + block - 1) / block;

    hipLaunchKernelGGL(lmht_lif_kernel, dim3(grid), dim3(block), 0, stream,
                       x0, x1, x2, x3, o0, o1, o2, o3, scale, zp, nvec);
}